// Prediction_46651934769530
// MI455X (gfx1250) — compile-verified
//
#include <hip/hip_runtime.h>
#include <math.h>

// Problem dimensions (fixed by reference setup_inputs)
#define B_    8
#define D_    512
#define N_    2048
#define KNN_  10
#define EPS_  1e-6f
#define P_EPS_ 1e-7f
#define SHARP_ 20.0f

typedef __attribute__((ext_vector_type(2))) float v2f;
typedef __attribute__((ext_vector_type(8))) float v8f;

// ---------------------------------------------------------------------------
// Stage A: fused GEMM (f32 WMMA 16x16x4) + online softmax(10000*scores/sqrt(D))
//          -> corr_tgt (B,N,3), corres (argmax index)
//
// One wave32 owns a 16-row n-strip and sweeps 128 m-tiles (two at a time so
// each A fragment feeds two WMMAs). Softmax state is kept LANE-LOCAL:
// lane l handles columns m == (l&15) mod 16, so the running max / argmax /
// sum-exp / exp-weighted tgt accumulators update branchlessly with no
// cross-lane traffic; the 16-lane merge happens once in the epilogue.
// ---------------------------------------------------------------------------
__global__ __launch_bounds__(256) void attn_softmax_kernel(
    const float* __restrict__ srcE, const float* __restrict__ tgtE,
    const float* __restrict__ tgt,
    float* __restrict__ corr_tgt, int* __restrict__ corres_i,
    float* __restrict__ corres_f)
{
  const int wave = threadIdx.x >> 5;
  const int lane = threadIdx.x & 31;
  const int tile = blockIdx.x * 8 + wave;          // 0..1023 strips
  const int b    = tile >> 7;                       // / (N_/16)
  const int nt   = tile & 127;
  const int n0   = nt * 16;
  const int l16  = lane & 15;
  const int half = lane >> 4;

  const float* Sb = srcE + (size_t)b * D_ * N_;
  const float* Tb = tgtE + (size_t)b * D_ * N_;
  const float* Pt = tgt  + (size_t)b * 3  * N_;

  const float scale = 10000.0f / sqrtf((float)D_);

  // per-lane online-softmax state for 8 rows (rows k and k+8 of the strip)
  float lmax[8], lsum[8], la0[8], la1[8], la2[8];
  int   larg[8];
#pragma unroll
  for (int k = 0; k < 8; ++k) {
    lmax[k] = -__builtin_inff();
    lsum[k] = 0.0f; la0[k] = 0.0f; la1[k] = 0.0f; la2[k] = 0.0f;
    larg[k] = 0;
  }

  const float* Abase = Sb + (size_t)(2 * half) * N_ + n0 + l16;
  const float* Bbase = Tb + (size_t)(2 * half) * N_ + l16;

  for (int mt = 0; mt < N_ / 16; mt += 2) {
    const int m0 = mt * 16;
    v8f c0 = {}, c1 = {};
    const float* Ap  = Abase;
    const float* Bp0 = Bbase + m0;
    const float* Bp1 = Bp0 + 16;
#pragma unroll 4
    for (int d = 0; d < D_; d += 4) {
#if defined(__HIP_DEVICE_COMPILE__) && __has_builtin(__builtin_amdgcn_wmma_f32_16x16x4_f32)
      v2f a, b0, b1;
      a.x  = Ap[0];    a.y  = Ap[N_];
      b0.x = Bp0[0];   b0.y = Bp0[N_];
      b1.x = Bp1[0];   b1.y = Bp1[N_];
      c0 = __builtin_amdgcn_wmma_f32_16x16x4_f32(
               false, a, false, b0, (short)0, c0, false, false);
      c1 = __builtin_amdgcn_wmma_f32_16x16x4_f32(
               false, a, false, b1, (short)0, c1, false, false);
#else
      // scalar fallback (kept only so host pass / builtin-less toolchains parse)
#pragma unroll
      for (int k = 0; k < 8; ++k) {
        const int row = n0 + k + 8 * half;
#pragma unroll
        for (int kk = 0; kk < 4; ++kk) {
          c0[k] += Sb[(size_t)(d + kk) * N_ + row] *
                   Tb[(size_t)(d + kk) * N_ + m0 + l16];
          c1[k] += Sb[(size_t)(d + kk) * N_ + row] *
                   Tb[(size_t)(d + kk) * N_ + m0 + 16 + l16];
        }
      }
#endif
      Ap  += 4 * N_;
      Bp0 += 4 * N_;
      Bp1 += 4 * N_;
    }
    // branchless lane-local softmax update for the two columns of this lane
#pragma unroll
    for (int half_t = 0; half_t < 2; ++half_t) {
      const int m = m0 + 16 * half_t + l16;
      const float t0 = Pt[m], t1 = Pt[N_ + m], t2 = Pt[2 * N_ + m];
#pragma unroll
      for (int k = 0; k < 8; ++k) {
        const float logit  = (half_t ? c1[k] : c0[k]) * scale;
        const float newmax = fmaxf(lmax[k], logit);
        const float corrf  = __expf(lmax[k] - newmax);  // exp(-inf)=0 first hit
        const float e      = __expf(logit - newmax);
        lsum[k] = lsum[k] * corrf + e;
        la0[k]  = la0[k]  * corrf + e * t0;
        la1[k]  = la1[k]  * corrf + e * t1;
        la2[k]  = la2[k]  * corrf + e * t2;
        larg[k] = (logit > lmax[k]) ? m : larg[k];  // keeps first occurrence
        lmax[k] = newmax;
      }
    }
  }

  // epilogue: merge the 16 lane-local states per row (branchless butterflies)
#pragma unroll
  for (int k = 0; k < 8; ++k) {
    float gmax = lmax[k];
#pragma unroll
    for (int off = 1; off < 16; off <<= 1) {
      const float o = __shfl_xor(gmax, off, 32);
      gmax = fmaxf(gmax, o);
    }
    const float sc = __expf(lmax[k] - gmax);
    float s  = lsum[k] * sc;
    float a0 = la0[k] * sc, a1 = la1[k] * sc, a2 = la2[k] * sc;
    int cand = (lmax[k] == gmax) ? larg[k] : 0x7fffffff;
#pragma unroll
    for (int off = 1; off < 16; off <<= 1) {
      s  += __shfl_xor(s,  off, 32);
      a0 += __shfl_xor(a0, off, 32);
      a1 += __shfl_xor(a1, off, 32);
      a2 += __shfl_xor(a2, off, 32);
      const int oc = __shfl_xor(cand, off, 32);
      cand = (oc < cand) ? oc : cand;
    }
    if (l16 == 0) {
      const int n = n0 + k + 8 * half;
      const size_t o = (size_t)b * N_ + n;
      const float inv = 1.0f / s;
      corr_tgt[o * 3 + 0] = a0 * inv;
      corr_tgt[o * 3 + 1] = a1 * inv;
      corr_tgt[o * 3 + 2] = a2 * inv;
      corres_i[o] = cand;
      corres_f[o] = (float)cand;
    }
  }
}

// ---------------------------------------------------------------------------
// Stage B1: 10-NN by w=exp(-d2/2) (descending w, lower index on ties == top_k)
// Point cloud (24KB) staged in LDS per block.
// ---------------------------------------------------------------------------
__global__ __launch_bounds__(256) void knn_kernel(
    const float* __restrict__ src, int* __restrict__ knn)
{
  __shared__ float px[N_], py[N_], pz[N_];
  const int b = blockIdx.x >> 3;          // 8 blocks per batch
  const int chunk = blockIdx.x & 7;
  const float* P = src + (size_t)b * 3 * N_;
  for (int i = threadIdx.x; i < N_; i += 256) {
    px[i] = P[i]; py[i] = P[N_ + i]; pz[i] = P[2 * N_ + i];
  }
  __syncthreads();
  const int n = chunk * 256 + threadIdx.x;
  const float xn = px[n], yn = py[n], zn = pz[n];
  const float sqn = xn * xn + yn * yn + zn * zn;
  float bw[KNN_]; int bi[KNN_];
#pragma unroll
  for (int k = 0; k < KNN_; ++k) { bw[k] = -1.0f; bi[k] = 0; }
  for (int m = 0; m < N_; ++m) {
    const float xm = px[m], ym = py[m], zm = pz[m];
    const float sqm = xm * xm + ym * ym + zm * zm;
    const float d2 = sqn + sqm - 2.0f * (xn * xm + yn * ym + zn * zm);
    const float w = __expf(-fmaxf(d2, 0.0f) * 0.5f);
    if (w > bw[KNN_ - 1]) {
      int pos = KNN_ - 1;
      while (pos > 0 && w > bw[pos - 1]) {
        bw[pos] = bw[pos - 1]; bi[pos] = bi[pos - 1]; --pos;
      }
      bw[pos] = w; bi[pos] = m;
    }
  }
  int* kn = knn + ((size_t)b * N_ + n) * KNN_;
#pragma unroll
  for (int k = 0; k < KNN_; ++k) kn[k] = bi[k];
}

// ---------------------------------------------------------------------------
// Stage B2: 45 triangles per point -> robust loss -> mean_loss (B,N)
// ---------------------------------------------------------------------------
__global__ __launch_bounds__(128) void tri_kernel(
    const float* __restrict__ src, const float* __restrict__ corr_tgt,
    const int* __restrict__ knn, float* __restrict__ mean_loss)
{
  const int gid = blockIdx.x * 128 + threadIdx.x;   // 0..16383
  const int b = gid >> 11, n = gid & (N_ - 1);
  const float* P = src + (size_t)b * 3 * N_;
  const float* Q = corr_tgt + (size_t)b * N_ * 3;
  const int* kn = knn + ((size_t)b * N_ + n) * KNN_;

  float sx[KNN_], sy[KNN_], sz[KNN_], tx[KNN_], ty[KNN_], tz[KNN_];
#pragma unroll
  for (int k = 0; k < KNN_; ++k) {
    const int i = kn[k];
    sx[k] = P[i]; sy[k] = P[N_ + i]; sz[k] = P[2 * N_ + i];
    tx[k] = Q[i * 3]; ty[k] = Q[i * 3 + 1]; tz[k] = Q[i * 3 + 2];
  }
  const float s0x = P[n], s0y = P[N_ + n], s0z = P[2 * N_ + n];
  const float t0x = Q[n * 3], t0y = Q[n * 3 + 1], t0z = Q[n * 3 + 2];

  float L[45], Rg[45];
  int t = 0;
  for (int i = 0; i < KNN_; ++i)
    for (int j = i + 1; j < KNN_; ++j) {
      // source triangle area (vertex0 = self)
      float e1x = sx[i] - s0x, e1y = sy[i] - s0y, e1z = sz[i] - s0z;
      float e2x = sx[j] - s0x, e2y = sy[j] - s0y, e2z = sz[j] - s0z;
      float cx = e1y * e2z - e1z * e2y, cy = e1z * e2x - e1x * e2z,
            cz = e1x * e2y - e1y * e2x;
      float as = 0.5f * sqrtf(cx * cx + cy * cy + cz * cz);
      // target triangle area
      float f1x = tx[i] - t0x, f1y = ty[i] - t0y, f1z = tz[i] - t0z;
      float f2x = tx[j] - t0x, f2y = ty[j] - t0y, f2z = tz[j] - t0z;
      float gx = f1y * f2z - f1z * f2y, gy = f1z * f2x - f1x * f2z,
            gz = f1x * f2y - f1y * f2x;
      float at = 0.5f * sqrtf(gx * gx + gy * gy + gz * gz);
      // sl=[0,0,as]; tl=[EPS,EPS,at+EPS]
      float diff = as - at - EPS_;
      float sumsq = 2.0f * EPS_ * EPS_ + diff * diff;
      L[t]  = sumsq / (as + at + 3.0f * EPS_);
      Rg[t] = sqrtf(sumsq);
      ++t;
    }
  // sort loss ascending (insertion)
  for (int i = 1; i < 45; ++i) {
    float key = L[i]; int j = i - 1;
    while (j >= 0 && L[j] > key) { L[j + 1] = L[j]; --j; }
    L[j + 1] = key;
  }
  // total = sorted_loss + 0.1*reg (reg in ORIGINAL triangle order, per reference)
  for (int i = 0; i < 45; ++i) L[i] = L[i] + 0.1f * Rg[i];
  // median of total: sort a copy (reuse Rg)
  for (int i = 0; i < 45; ++i) Rg[i] = L[i];
  for (int i = 1; i < 45; ++i) {
    float key = Rg[i]; int j = i - 1;
    while (j >= 0 && Rg[j] > key) { Rg[j + 1] = Rg[j]; --j; }
    Rg[j + 1] = key;
  }
  const float thr = 3.0f * Rg[22];
  float acc = 0.0f;
#pragma unroll
  for (int i = 0; i < KNN_; ++i) {
    float tv = L[i];
    tv = (tv > thr) ? 0.0f : tv;
    acc += sqrtf(tv + EPS_);
  }
  mean_loss[(size_t)b * N_ + n] = acc * (1.0f / KNN_);
}

// ---------------------------------------------------------------------------
// Stage B3: per-batch min of mean_loss
// ---------------------------------------------------------------------------
__global__ __launch_bounds__(256) void minred_kernel(
    const float* __restrict__ mean_loss, float* __restrict__ min_loss)
{
  __shared__ float red[256];
  const int b = blockIdx.x;
  float m = __builtin_inff();
  for (int i = threadIdx.x; i < N_; i += 256)
    m = fminf(m, mean_loss[(size_t)b * N_ + i]);
  red[threadIdx.x] = m;
  __syncthreads();
  for (int s = 128; s > 0; s >>= 1) {
    if (threadIdx.x < s) red[threadIdx.x] = fminf(red[threadIdx.x], red[threadIdx.x + s]);
    __syncthreads();
  }
  if (threadIdx.x == 0) min_loss[b] = red[0];
}

// ---------------------------------------------------------------------------
// Stage B4: weight = (2*sigmoid(-SHARP*(ml-min)) > 0.5) ? 1 : 0
// ---------------------------------------------------------------------------
__global__ __launch_bounds__(256) void weight_kernel(
    const float* __restrict__ mean_loss, const float* __restrict__ min_loss,
    float* __restrict__ weight_out)
{
  const int gid = blockIdx.x * 256 + threadIdx.x;
  const int b = gid >> 11;
  const float d = mean_loss[gid] - min_loss[b];
  const float w = 2.0f / (1.0f + __expf(SHARP_ * d));
  weight_out[gid] = (w > 0.5f) ? 1.0f : 0.0f;
}

// ---------------------------------------------------------------------------
// Stage C: weighted Procrustes per batch (3x3 Jacobi SVD on thread 0)
// ---------------------------------------------------------------------------
__device__ static inline float det3(const float M[3][3]) {
  return M[0][0] * (M[1][1] * M[2][2] - M[1][2] * M[2][1])
       - M[0][1] * (M[1][0] * M[2][2] - M[1][2] * M[2][0])
       + M[0][2] * (M[1][0] * M[2][1] - M[1][1] * M[2][0]);
}

__device__ static void svd_rotation(const float Am[3][3], float R[3][3]) {
  float S[3][3], V[3][3];
#pragma unroll
  for (int a = 0; a < 3; ++a)
#pragma unroll
    for (int c = 0; c < 3; ++c)
      S[a][c] = Am[0][a] * Am[0][c] + Am[1][a] * Am[1][c] + Am[2][a] * Am[2][c];
#pragma unroll
  for (int a = 0; a < 3; ++a)
#pragma unroll
    for (int c = 0; c < 3; ++c) V[a][c] = (a == c) ? 1.0f : 0.0f;
  const int PI_[3] = {0, 0, 1}, QI_[3] = {1, 2, 2};
  for (int it = 0; it < 30; ++it) {
    const int p = PI_[it % 3], q = QI_[it % 3];
    const float apq = S[p][q];
    if (fabsf(apq) > 1e-30f) {
      const float tau = (S[q][q] - S[p][p]) / (2.0f * apq);
      const float tt = (tau >= 0.0f) ? 1.0f / (tau + sqrtf(1.0f + tau * tau))
                                     : 1.0f / (tau - sqrtf(1.0f + tau * tau));
      const float c0 = 1.0f / sqrtf(1.0f + tt * tt);
      const float s0 = tt * c0;
      for (int k = 0; k < 3; ++k) { float kp = S[k][p], kq = S[k][q];
        S[k][p] = c0 * kp - s0 * kq; S[k][q] = s0 * kp + c0 * kq; }
      for (int k = 0; k < 3; ++k) { float pk = S[p][k], qk = S[q][k];
        S[p][k] = c0 * pk - s0 * qk; S[q][k] = s0 * pk + c0 * qk; }
      for (int k = 0; k < 3; ++k) { float vp = V[k][p], vq = V[k][q];
        V[k][p] = c0 * vp - s0 * vq; V[k][q] = s0 * vp + c0 * vq; }
    }
  }
  float ev[3] = {S[0][0], S[1][1], S[2][2]};
  for (int i = 0; i < 2; ++i)
    for (int j = 0; j < 2 - i; ++j)
      if (ev[j] < ev[j + 1]) {
        float tv = ev[j]; ev[j] = ev[j + 1]; ev[j + 1] = tv;
        for (int k = 0; k < 3; ++k) { float t2 = V[k][j]; V[k][j] = V[k][j + 1]; V[k][j + 1] = t2; }
      }
  float u0[3], u1[3], u2[3];
  for (int k = 0; k < 3; ++k)
    u0[k] = Am[k][0] * V[0][0] + Am[k][1] * V[1][0] + Am[k][2] * V[2][0];
  for (int k = 0; k < 3; ++k)
    u1[k] = Am[k][0] * V[0][1] + Am[k][1] * V[1][1] + Am[k][2] * V[2][1];
  const float n0 = sqrtf(u0[0] * u0[0] + u0[1] * u0[1] + u0[2] * u0[2]);
  const float n1 = sqrtf(u1[0] * u1[0] + u1[1] * u1[1] + u1[2] * u1[2]);
  const float i0 = (n0 > 1e-25f) ? 1.0f / n0 : 0.0f;
  const float i1 = (n1 > 1e-25f) ? 1.0f / n1 : 0.0f;
  for (int k = 0; k < 3; ++k) { u0[k] *= i0; u1[k] *= i1; }
  // u2 = u0 x u1: sign*u2*v2^T is invariant to the SVD sign ambiguity of u2
  u2[0] = u0[1] * u1[2] - u0[2] * u1[1];
  u2[1] = u0[2] * u1[0] - u0[0] * u1[2];
  u2[2] = u0[0] * u1[1] - u0[1] * u1[0];
  const float sgn = (det3(V) < 0.0f) ? -1.0f : 1.0f;  // det(U)=+1 by construction
#pragma unroll
  for (int a = 0; a < 3; ++a)
#pragma unroll
    for (int c = 0; c < 3; ++c)
      R[a][c] = u0[a] * V[c][0] + u1[a] * V[c][1] + sgn * u2[a] * V[c][2];
}

__device__ static inline float block_sum(float v, float* red) {
  red[threadIdx.x] = v;
  __syncthreads();
  for (int s = 128; s > 0; s >>= 1) {
    if (threadIdx.x < s) red[threadIdx.x] += red[threadIdx.x + s];
    __syncthreads();
  }
  const float r = red[0];
  __syncthreads();
  return r;
}

__global__ __launch_bounds__(256) void procrustes_kernel(
    const float* __restrict__ src, const float* __restrict__ tgt,
    const int* __restrict__ corres_i, const float* __restrict__ weight,
    float* __restrict__ R_out, float* __restrict__ T_out)
{
  __shared__ float red[256];
  const int b = blockIdx.x;
  const float* X = src + (size_t)b * 3 * N_;
  const float* Y = tgt + (size_t)b * 3 * N_;
  const int* ci = corres_i + (size_t)b * N_;
  const float* w = weight + (size_t)b * N_;

  float lw = 0.0f, lx0 = 0, lx1 = 0, lx2 = 0, ly0 = 0, ly1 = 0, ly2 = 0;
  for (int n = threadIdx.x; n < N_; n += 256) {
    const float wn = w[n];
    const int c = ci[n];
    lw += wn;
    lx0 += wn * X[n];      lx1 += wn * X[N_ + n];  lx2 += wn * X[2 * N_ + n];
    ly0 += wn * Y[c];      ly1 += wn * Y[N_ + c];  ly2 += wn * Y[2 * N_ + c];
  }
  const float Sw = block_sum(lw, red);
  const float inv = 1.0f / (Sw + P_EPS_);
  float mx[3], my[3];
  mx[0] = block_sum(lx0, red) * inv; mx[1] = block_sum(lx1, red) * inv;
  mx[2] = block_sum(lx2, red) * inv;
  my[0] = block_sum(ly0, red) * inv; my[1] = block_sum(ly1, red) * inv;
  my[2] = block_sum(ly2, red) * inv;

  float lc[9];
#pragma unroll
  for (int i = 0; i < 9; ++i) lc[i] = 0.0f;
  for (int n = threadIdx.x; n < N_; n += 256) {
    const float wv = w[n] * inv;
    const int c = ci[n];
    const float xc0 = X[n] - mx[0], xc1 = X[N_ + n] - mx[1], xc2 = X[2 * N_ + n] - mx[2];
    const float yc0 = Y[c] - my[0], yc1 = Y[N_ + c] - my[1], yc2 = Y[2 * N_ + c] - my[2];
    lc[0] += wv * yc0 * xc0; lc[1] += wv * yc0 * xc1; lc[2] += wv * yc0 * xc2;
    lc[3] += wv * yc1 * xc0; lc[4] += wv * yc1 * xc1; lc[5] += wv * yc1 * xc2;
    lc[6] += wv * yc2 * xc0; lc[7] += wv * yc2 * xc1; lc[8] += wv * yc2 * xc2;
  }
  float cov[3][3];
  for (int i = 0; i < 9; ++i) cov[i / 3][i % 3] = block_sum(lc[i], red);

  if (threadIdx.x == 0) {
    float R[3][3];
    svd_rotation(cov, R);
    float Rm[3];
#pragma unroll
    for (int i = 0; i < 3; ++i)
      Rm[i] = R[i][0] * mx[0] + R[i][1] * mx[1] + R[i][2] * mx[2];
    float* Rb = R_out + b * 9;
    float* Tb = T_out + b * 9;   // T broadcasts to (3,3) in the reference!
#pragma unroll
    for (int i = 0; i < 3; ++i)
#pragma unroll
      for (int j = 0; j < 3; ++j) {
        Rb[i * 3 + j] = R[i][j];
        Tb[i * 3 + j] = my[j] - Rm[i];
      }
  }
}

// ---------------------------------------------------------------------------
// Host-side launcher
// ---------------------------------------------------------------------------
extern "C" void kernel_launch(void* const* d_in, const int* in_sizes, int n_in,
                              void* d_out, int out_size, void* d_ws, size_t ws_size,
                              hipStream_t stream) {
  (void)in_sizes; (void)n_in; (void)out_size; (void)ws_size;
  const float* srcE = (const float*)d_in[0];
  const float* tgtE = (const float*)d_in[1];
  const float* src  = (const float*)d_in[2];
  const float* tgt  = (const float*)d_in[3];
  // d_in[4]=temperature, d_in[5]=is_corr: unused by the reference computation

  float* out       = (float*)d_out;
  float* R_out     = out;                       // (8,3,3)  = 72
  float* T_out     = out + 72;                  // (8,3,3)  = 72 (broadcasted T)
  float* corres_f  = out + 144;                 // (8,2048,1)
  float* weight_o  = out + 144 + B_ * N_;       // (8,2048,1)

  float* wsf       = (float*)d_ws;
  float* corr_tgt  = wsf;                                   // 49152 f32
  int*   corres_i  = (int*)(wsf + B_ * N_ * 3);             // 16384 i32
  int*   knn       = (int*)(wsf + B_ * N_ * 3 + B_ * N_);   // 163840 i32
  float* mean_loss = wsf + B_ * N_ * 3 + B_ * N_ + B_ * N_ * KNN_;  // 16384 f32
  float* min_loss  = mean_loss + B_ * N_;                   // 8 f32

  // A: GEMM + online softmax (1024 wave-strips, 8 waves/block)
  attn_softmax_kernel<<<128, 256, 0, stream>>>(srcE, tgtE, tgt,
                                               corr_tgt, corres_i, corres_f);
  // B1: 10-NN over src points
  knn_kernel<<<64, 256, 0, stream>>>(src, knn);
  // B2: triangle losses -> mean_loss
  tri_kernel<<<128, 128, 0, stream>>>(src, corr_tgt, knn, mean_loss);
  // B3: per-batch min
  minred_kernel<<<8, 256, 0, stream>>>(mean_loss, min_loss);
  // B4: binary weights
  weight_kernel<<<64, 256, 0, stream>>>(mean_loss, min_loss, weight_o);
  // C: weighted Procrustes -> R, T
  procrustes_kernel<<<8, 256, 0, stream>>>(src, tgt, corres_i, weight_o,
                                           R_out, T_out);
}